// GATLayer_39805756899551
// MI455X (gfx1250) — compile-verified
//
#include <hip/hip_runtime.h>
#include <math.h>

#define NN 50000          // N_NODES
#define NE 1600000        // N_EDGES
#define DIN 128
#define DOUT 32
#define NH 4
#define DT 128            // NH * DOUT
#define NTILES (NN / 16)  // 3125, exact
#define NPAIR (DIN / 2)   // 64 K-pair rows
#define WPAD 144          // pairs per row: 288 dwords ≡ 32 (mod 64) -> halves hit disjoint banks

typedef __attribute__((ext_vector_type(2))) float v2f;
typedef __attribute__((ext_vector_type(8))) float v8f;

// ---------------- Phase 1: Wh = x @ W  via V_WMMA_F32_16X16X4_F32 ----------------
// W staged per-block into LDS as packed K-pairs: Wsh[p][j] = (B[2p][j], B[2p+1][j]),
// j = head*32+o. Each lane's B fragment is then ONE aligned ds_load_b64 directly
// into an even VGPR pair (no VALU packing), while x streams through VMEM b64 loads.
// One wave per 16-node row tile; 8 column tiles x 32 K-steps (K=4 each).
__global__ void __launch_bounds__(128) gat_wh_gemm(const float* __restrict__ x,
                                                   const float* __restrict__ W,
                                                   float* __restrict__ Wh) {
  __shared__ v2f Wsh[NPAIR * WPAD];   // 72 KB of the 320 KB WGP LDS

  // Cooperative staging of the packed-pair layout
  for (int idx = threadIdx.x; idx < NPAIR * DT; idx += 128) {
    const int p  = idx >> 7;          // K-pair row: input rows i = 2p, 2p+1
    const int j  = idx & 127;         // output column
    const int kh = j >> 5;
    const int o  = j & 31;
    const float* wp = W + kh * (DIN * DOUT) + (2 * p) * DOUT + o;
    v2f pr;
    pr.x = wp[0];
    pr.y = wp[DOUT];
    Wsh[p * WPAD + j] = pr;
  }
  __syncthreads();

  const int lane = threadIdx.x & 31;
  const int wave = threadIdx.x >> 5;
  const int half = lane >> 4;   // 0: lanes 0-15, 1: lanes 16-31
  const int l16  = lane & 15;
  const int tile = blockIdx.x * 4 + wave;
  if (tile >= NTILES) return;   // wave-uniform: EXEC stays all-1s for active waves
  const int row0 = tile * 16;

  v8f acc[8];
#pragma unroll
  for (int jt = 0; jt < 8; ++jt) acc[jt] = (v8f){0.f,0.f,0.f,0.f,0.f,0.f,0.f,0.f};

  // A fragment rows: lane l16 holds row row0+l16; half selects K-pair within 4-wide step
  const float* xrow = x + (size_t)(row0 + l16) * DIN;

#pragma unroll 4
  for (int kk = 0; kk < 32; ++kk) {
    const int i0 = kk * 4 + half * 2;               // this lane's two K indices: i0, i0+1
    const int p  = kk * 2 + half;                   // K-pair row in LDS
    const v2f a = *(const v2f*)(xrow + i0);         // global_load_b64 (8B aligned)
    const v2f* wr = Wsh + p * WPAD + l16;
#pragma unroll
    for (int jt = 0; jt < 8; ++jt) {
      const v2f b = wr[jt * 16];                    // ds_load_b64, conflict-free
      acc[jt] = __builtin_amdgcn_wmma_f32_16x16x4_f32(
          false, a, false, b, (short)0, acc[jt], false, false);
    }
  }

  // C/D layout: VGPR r -> row r + 8*half, col = l16 within each 16-wide tile
#pragma unroll
  for (int jt = 0; jt < 8; ++jt) {
#pragma unroll
    for (int r = 0; r < 8; ++r) {
      const int row = row0 + r + 8 * half;
      Wh[(size_t)row * DT + jt * 16 + l16] = acc[jt][r];
    }
  }
}

// ---------------- init accumulators + output ----------------
__global__ void gat_init(float* __restrict__ out, float* __restrict__ e_max,
                         float* __restrict__ e_sum) {
  int i = blockIdx.x * blockDim.x + threadIdx.x;
  if (i < NN * DT) out[i] = 0.f;
  if (i < NN * NH) { e_sum[i] = 0.f; e_max[i] = -__builtin_inff(); }
}

// ---------------- per-node attention scores s_src, s_tgt ----------------
__global__ void gat_scores(const float* __restrict__ Wh, const float* __restrict__ a,
                           float* __restrict__ s_src, float* __restrict__ s_tgt) {
  int idx = blockIdx.x * blockDim.x + threadIdx.x;
  if (idx >= NN * NH) return;
  int n = idx >> 2, k = idx & 3;
  const float* w  = Wh + (size_t)n * DT + k * DOUT;
  const float* as = a + k * (2 * DOUT);
  const float* at = as + DOUT;
  float ss = 0.f, st = 0.f;
#pragma unroll
  for (int o = 0; o < DOUT; ++o) { float v = w[o]; ss += v * as[o]; st += v * at[o]; }
  s_src[idx] = ss;
  s_tgt[idx] = st;
}

// float atomic max (handles mixed signs + -inf init) via signed-max / unsigned-min
__device__ inline void atomicMaxF(float* addr, float val) {
  if (val >= 0.f) atomicMax((int*)addr, __float_as_int(val));
  else            atomicMin((unsigned int*)addr, __float_as_uint(val));
}

// ---------------- edge pass A: e = LeakyReLU(s_src[src]+s_tgt[tgt]); segment max ----------------
__global__ void gat_edge_max(const int* __restrict__ src, const int* __restrict__ tgt,
                             const float* __restrict__ s_src, const float* __restrict__ s_tgt,
                             float* __restrict__ e_buf, float* __restrict__ e_max) {
  int idx = blockIdx.x * blockDim.x + threadIdx.x;
  if (idx >= NE * NH) return;
  int e = idx >> 2, k = idx & 3;
  int s = src[e], t = tgt[e];
  float v = s_src[s * NH + k] + s_tgt[t * NH + k];
  v = v > 0.f ? v : 0.2f * v;     // LeakyReLU(0.2)
  e_buf[idx] = v;
  atomicMaxF(&e_max[t * NH + k], v);
}

// ---------------- edge pass B: e_exp = exp(e - max); segment sum ----------------
__global__ void gat_edge_exp(const int* __restrict__ tgt, const float* __restrict__ e_max,
                             float* __restrict__ e_buf, float* __restrict__ e_sum) {
  int idx = blockIdx.x * blockDim.x + threadIdx.x;
  if (idx >= NE * NH) return;
  int e = idx >> 2, k = idx & 3;
  int t = tgt[e];
  float ex = __expf(e_buf[idx] - e_max[t * NH + k]);
  e_buf[idx] = ex;
  atomicAdd(&e_sum[t * NH + k], ex);
}

// ---------------- edge pass C: out[tgt] += alpha * Wh[src]  (coalesced over j) ----------------
__global__ void gat_edge_agg(const int* __restrict__ src, const int* __restrict__ tgt,
                             const float* __restrict__ e_buf, const float* __restrict__ e_sum,
                             const float* __restrict__ Wh, float* __restrict__ out) {
  long long idx = (long long)blockIdx.x * blockDim.x + threadIdx.x;
  if (idx >= (long long)NE * DT) return;
  int e = (int)(idx >> 7);
  int j = (int)(idx & 127);
  int k = j >> 5;
  int s = src[e], t = tgt[e];
  float alpha = e_buf[(size_t)e * NH + k] / (e_sum[t * NH + k] + 1e-10f);
  atomicAdd(&out[(size_t)t * DT + j], alpha * Wh[(size_t)s * DT + j]);
}

// ---------------- final ELU (in place on d_out) ----------------
__global__ void gat_elu(float* __restrict__ out) {
  int i = blockIdx.x * blockDim.x + threadIdx.x;
  if (i >= NN * DT) return;
  float v = out[i];
  out[i] = v > 0.f ? v : expm1f(v);
}

extern "C" void kernel_launch(void* const* d_in, const int* in_sizes, int n_in,
                              void* d_out, int out_size, void* d_ws, size_t ws_size,
                              hipStream_t stream) {
  const float* x    = (const float*)d_in[0];
  const int*   eidx = (const int*)d_in[1];   // [2, NE]
  const float* W    = (const float*)d_in[2]; // [NH, DIN, DOUT]
  const float* a    = (const float*)d_in[3]; // [NH, 2*DOUT]
  float* out = (float*)d_out;

  const int* src = eidx;        // edge_index[0]
  const int* tgt = eidx + NE;   // edge_index[1]

  // workspace layout (floats): Wh | s_src | s_tgt | e_max | e_sum | e_buf  (~54.4 MB)
  float* ws    = (float*)d_ws;
  float* Wh    = ws;
  float* s_src = Wh + (size_t)NN * DT;
  float* s_tgt = s_src + NN * NH;
  float* e_max = s_tgt + NN * NH;
  float* e_sum = e_max + NN * NH;
  float* e_buf = e_sum + NN * NH;

  const int n_out = NN * DT;
  gat_init<<<(n_out + 255) / 256, 256, 0, stream>>>(out, e_max, e_sum);

  // WMMA GEMM: 4 waves/block, 1 row-tile per wave, W staged in LDS (packed pairs)
  gat_wh_gemm<<<(NTILES + 3) / 4, 128, 0, stream>>>(x, W, Wh);

  gat_scores<<<(NN * NH + 255) / 256, 256, 0, stream>>>(Wh, a, s_src, s_tgt);

  const int ne4 = NE * NH;
  gat_edge_max<<<(ne4 + 255) / 256, 256, 0, stream>>>(src, tgt, s_src, s_tgt, e_buf, e_max);
  gat_edge_exp<<<(ne4 + 255) / 256, 256, 0, stream>>>(tgt, e_max, e_buf, e_sum);

  const long long neagg = (long long)NE * DT;
  gat_edge_agg<<<(int)((neagg + 255) / 256), 256, 0, stream>>>(src, tgt, e_buf, e_sum, Wh, out);

  gat_elu<<<(n_out + 255) / 256, 256, 0, stream>>>(out);
}